// PhoenixEmulator_12876311953883
// MI455X (gfx1250) — compile-verified
//
#include <hip/hip_runtime.h>
#include <hip/hip_bf16.h>

#define N_LINES 2048
#define N_PIX   131072
#define PSTRIDE 6                  // floats per line record (5 used + 1 pad)
#define BLOCK   256
#define PXT     2                  // pixels per thread
#define LOG2E   1.4426950408889634f

typedef unsigned int u32x4 __attribute__((ext_vector_type(4)));
typedef int          i32x8 __attribute__((ext_vector_type(8)));
typedef int          i32x4 __attribute__((ext_vector_type(4)));
typedef float        f32x2 __attribute__((ext_vector_type(2)));

#if defined(__gfx1250__) && __has_builtin(__builtin_amdgcn_tensor_load_to_lds)
#define USE_TDM 1
#else
#define USE_TDM 0
#endif

// ---------------------------------------------------------------------------
// Kernel 1: per-line parameter preparation (2048 threads; negligible cost).
//   p0 = lam
//   p1 = gam^2
//   p2 = amp*eta*gam/pi                  (Lorentz coefficient)
//   p3 = -0.5*log2(e)/sig^2              (Gauss exponent scale, base-2)
//   p4 = amp*(1-eta)/(sig*2.5066)        (Gauss coefficient)
// ---------------------------------------------------------------------------
__global__ void voigt_prep(const float* __restrict__ lam_c,
                           const float* __restrict__ log_amps,
                           const float* __restrict__ log_sigmas,
                           const float* __restrict__ log_gammas,
                           float* __restrict__ lp)
{
    int i = blockIdx.x * blockDim.x + threadIdx.x;
    if (i >= N_LINES) return;

    float sig = __expf(log_sigmas[i]);
    float gam = __expf(log_gammas[i]);
    float amp = __expf(log_amps[i]);

    float fg  = 2.3548f * sig;
    float fl  = 2.0f * gam;
    float fg2 = fg * fg, fg3 = fg2 * fg, fg4 = fg2 * fg2, fg5 = fg4 * fg;
    float fl2 = fl * fl, fl3 = fl2 * fl, fl4 = fl2 * fl2, fl5 = fl4 * fl;
    float s = fg5 + 2.69269f * fg4 * fl + 2.42843f * fg3 * fl2 +
              4.47163f * fg2 * fl3 + 0.07842f * fg * fl4 + fl5;
    float fwhm = powf(s, 0.2f);
    float r    = fl / fwhm;
    float eta  = r * (1.36603f + r * (-0.47719f + r * 0.11116f));

    float* p = lp + i * PSTRIDE;
    p[0] = lam_c[i];
    p[1] = gam * gam;
    p[2] = amp * eta * gam * (1.0f / 3.141592654f);
    p[3] = -0.5f * LOG2E / (sig * sig);
    p[4] = amp * (1.0f - eta) / (sig * 2.5066f);
    p[5] = 0.0f;
}

// ---------------------------------------------------------------------------
// Kernel 2: main synthesis. One TDM DMA stages the 48KB line-param table into
// LDS per workgroup; each thread accumulates prod(1-profile) for 2 pixels,
// with the 2-pixel lanes expressed as float2 so the backend can use
// V_PK_*_F32 / VOPD dual-issue for everything except the trans ops.
// ---------------------------------------------------------------------------
__global__ void __launch_bounds__(BLOCK)
voigt_main(const float* __restrict__ wl,
           const float* __restrict__ lp,
           const float* __restrict__ a_c,
           const float* __restrict__ b_c,
           const float* __restrict__ c_c,
           float* __restrict__ out)
{
    __shared__ __align__(16) float sP[N_LINES * PSTRIDE];   // 48 KB

#if USE_TDM
    if (threadIdx.x < 32) {
        // ---- Tensor DMA descriptor: 1D copy of N_LINES*PSTRIDE dwords ----
        const unsigned nelem   = (unsigned)(N_LINES * PSTRIDE);   // 12288 dwords
        unsigned long long gaddr = (unsigned long long)(const void*)lp;
        unsigned laddr = (unsigned)(unsigned long long)(const void*)&sP[0]; // LDS byte offset (addr[31:0])

        u32x4 g0;
        g0.x = 1u;                                   // count=1 (valid user descriptor)
        g0.y = laddr;                                // lds_addr[31:0]
        g0.z = (unsigned)(gaddr & 0xFFFFFFFFull);    // global_addr[31:0]
        g0.w = (unsigned)((gaddr >> 32) & 0x1FFFFFFull) | (2u << 30); // global_addr[56:32] | type=2

        i32x8 g1;
        g1[0] = (int)(2u << 16);                     // workgroup_mask=0, data_size=2 (4B)
        g1[1] = (int)((nelem & 0xFFFFu) << 16);      // tensor_dim0[15:0]  (abar_addr=0)
        g1[2] = (int)((1u & 0xFFFFu) << 16) | (int)(nelem >> 16); // tensor_dim1=1 | tensor_dim0[31:16]
        g1[3] = (int)((nelem & 0xFFFFu) << 16);      // tile_dim0 = nelem | tensor_dim1[31:16]=0
        g1[4] = 1;                                   // tile_dim1=1, tile_dim2=0
        g1[5] = (int)nelem;                          // tensor_dim0_stride[31:0]
        g1[6] = 0;                                   // stride hi bits / tensor_dim1_stride lo
        g1[7] = 0;
        i32x4 g2 = {0, 0, 0, 0};
        i32x4 g3 = {0, 0, 0, 0};
        i32x8 g4 = {0, 0, 0, 0, 0, 0, 0, 0};

        __builtin_amdgcn_tensor_load_to_lds(g0, g1, g2, g3, g4, 0);
#if __has_builtin(__builtin_amdgcn_s_wait_tensorcnt)
        __builtin_amdgcn_s_wait_tensorcnt(0);
#else
        asm volatile("s_wait_tensorcnt 0x0" ::: "memory");
#endif
    }
    __syncthreads();
#else
    {
        const float4* __restrict__ src = (const float4*)lp;
        float4* dst = (float4*)sP;
        for (int i = threadIdx.x; i < (N_LINES * PSTRIDE) / 4; i += BLOCK)
            dst[i] = src[i];
        __syncthreads();
    }
#endif

    const int p0 = blockIdx.x * (BLOCK * PXT) + threadIdx.x;   // coalesced: thread t -> p0, p0+BLOCK
    f32x2 w;
    w.x = wl[p0];
    w.y = wl[p0 + BLOCK];

    f32x2 acc = {1.0f, 1.0f};

#pragma unroll 4
    for (int l = 0; l < N_LINES; ++l) {
        const float* __restrict__ p = &sP[l * PSTRIDE];
        const float lam  = p[0];
        const float gam2 = p[1];
        const float cl   = p[2];
        const float ks   = p[3];
        const float cg   = p[4];

        const f32x2 d   = w - lam;                                   // pk_add (neg)
        const f32x2 q   = d * d;                                     // pk_mul
        const f32x2 den = __builtin_elementwise_fma(d, d, (f32x2)(gam2)); // pk_fma
        const f32x2 arg = q * ks;                                    // pk_mul

        f32x2 r, e;
        r.x = __builtin_amdgcn_rcpf(den.x);                          // v_rcp_f32
        r.y = __builtin_amdgcn_rcpf(den.y);
        e.x = __builtin_amdgcn_exp2f(arg.x);                         // v_exp_f32
        e.y = __builtin_amdgcn_exp2f(arg.y);

        const f32x2 t    = e * cg;                                   // pk_mul
        const f32x2 prof = __builtin_elementwise_fma((f32x2)(cl), r, t); // pk_fma
        acc = __builtin_elementwise_fma(-prof, acc, acc);            // acc *= (1 - prof)
    }

    const float A = *a_c, B = *b_c, C = *c_c;
    const f32x2 xn = (w - 10500.0f) * (1.0f / 2500.0f);
    const f32x2 mod = __builtin_elementwise_fma(
        xn, __builtin_elementwise_fma(xn, (f32x2)(C), (f32x2)(B)), (f32x2)(A));
    const f32x2 res = acc * mod;
    out[p0]         = res.x;
    out[p0 + BLOCK] = res.y;
}

// ---------------------------------------------------------------------------
// Launcher
// ---------------------------------------------------------------------------
extern "C" void kernel_launch(void* const* d_in, const int* in_sizes, int n_in,
                              void* d_out, int out_size, void* d_ws, size_t ws_size,
                              hipStream_t stream)
{
    const float* wl          = (const float*)d_in[0];
    const float* lam_centers = (const float*)d_in[1];
    const float* log_amps    = (const float*)d_in[2];
    const float* log_sigmas  = (const float*)d_in[3];
    const float* log_gammas  = (const float*)d_in[4];
    const float* a_coeff     = (const float*)d_in[5];
    const float* b_coeff     = (const float*)d_in[6];
    const float* c_coeff     = (const float*)d_in[7];
    float* out = (float*)d_out;

    float* lp = (float*)d_ws;   // N_LINES * PSTRIDE floats = 48 KB

    voigt_prep<<<(N_LINES + 255) / 256, 256, 0, stream>>>(
        lam_centers, log_amps, log_sigmas, log_gammas, lp);

    voigt_main<<<N_PIX / (BLOCK * PXT), BLOCK, 0, stream>>>(
        wl, lp, a_coeff, b_coeff, c_coeff, out);
}